// EpisodicMemory_3075196584328
// MI455X (gfx1250) — compile-verified
//
#include <hip/hip_runtime.h>
#include <math.h>
#include <stdint.h>

// ---------------------------------------------------------------------------
// EpisodicMemory for MI455X (gfx1250, wave32).
// Memory-bound problem (~0.8 GFLOP vs ~670MB HBM traffic): strategy is to
// read em_K exactly once (fused 9-row score WMMA GEMM, async-staged to LDS),
// bulk-stream the K/V copies, and reduce the sequential 8-step write scan to
// sparse updates of <=32 slots per batch (unit_normalize is idempotent on
// untouched slots). All GEMMs use V_WMMA_F32_16X16X4_F32 (exact fp32, safe
// for top-k tie-breaking).
// ---------------------------------------------------------------------------

typedef __attribute__((ext_vector_type(2))) float v2f;
typedef __attribute__((ext_vector_type(8))) float v8f;
typedef __attribute__((ext_vector_type(4))) int v4i;

#define WMMA_F32(a, b, c) \
  __builtin_amdgcn_wmma_f32_16x16x4_f32(false, (a), false, (b), (short)0, (c), false, false)

#define NEG_INF (-__builtin_inff())

#if __has_builtin(__builtin_amdgcn_global_load_async_to_lds_b128)
#define HAVE_ASYNC_LDS 1
#else
#define HAVE_ASYNC_LDS 0
#endif

// addrspace casts matching LLVM's flat->AS lowering (AS3 = low 32 bits).
// The async-to-LDS builtin takes v4i pointers (per hipcc's diagnostic).
#define TO_GLOBAL(p) ((__attribute__((address_space(1))) v4i*)(uintptr_t)(p))
#define TO_LDS(p) ((__attribute__((address_space(3))) v4i*)(uint32_t)(uintptr_t)(p))

constexpr int kM = 4096;     // memory slots
constexpr int kD = 2048;     // hidden dim
constexpr int kDem = 256;    // embedding dim
constexpr int kBS = 32;      // batch
constexpr int kP = 512;      // candidates
constexpr int kKRET = 16;
constexpr int kC = 8;
constexpr int kKW = 4;
constexpr float kTAU = 0.5f;
constexpr float kWW = 0.1f;
constexpr float kSMAX = 10.0f;
constexpr float kBUDGET = 2048.0f;
constexpr float kDECAY = 0.99f;
constexpr float kEPS = 1e-12f;
constexpr float kCROSS = 0.0625f;  // 256^-0.5

// ---------------------------------------------------------------------------
// WMMA k-panel: acc += A_panel @ B_panel with incrementing pointers.
// ap: lane's A element ptr (A[arow, koff]), advances 4 floats/step.
// bp: lane's B element ptr (B[koff, n0+col]), advances 4 rows/step.
// ---------------------------------------------------------------------------
__device__ __forceinline__ void gemm_panel(const float*& ap, const float*& bp,
                                           int iters, int ldb, v8f& acc) {
  for (int i = 0; i < iters; ++i) {
    v2f a = *(const v2f*)ap;
    v2f bf;
    bf.x = bp[0];
    bf.y = bp[ldb];
    acc = WMMA_F32(a, bf, acc);
    ap += 4;
    bp += (size_t)4 * ldb;
  }
}

// ---------------------------------------------------------------------------
// Generic f32 WMMA GEMM: C[M,N] = A[M,K] @ B[K,N] (+bias), A split over two
// row-major pointers at column `ksplit` (for concat(x, y_wm)). One wave per
// 16x16 tile; K multiple of 4; M,N multiples of 16 in all uses here.
// ---------------------------------------------------------------------------
__global__ void __launch_bounds__(32)
wmma_gemm_bias(const float* __restrict__ A1, const float* __restrict__ A2, int ksplit,
               int lda, const float* __restrict__ B, const float* __restrict__ bias,
               float* __restrict__ C, int ldc, int M, int N, int K) {
  const int lane = threadIdx.x;
  const int n0 = blockIdx.x * 16, m0 = blockIdx.y * 16;
  const int col = lane & 15;
  const int koff = (lane & 16) ? 2 : 0;
  const int arow = m0 + col;

  union { v8f v; float f[8]; } acc;
#pragma unroll
  for (int i = 0; i < 8; i++) acc.f[i] = 0.0f;

  const float* bp = B + (size_t)koff * N + n0 + col;
  const float* ap = A1 + (size_t)arow * lda + koff;
  gemm_panel(ap, bp, ksplit / 4, N, acc.v);           // A1 panel
  if (K > ksplit) {
    ap = A2 + (size_t)arow * lda + koff;
    gemm_panel(ap, bp, (K - ksplit) / 4, N, acc.v);   // A2 panel (bp continues)
  }

#pragma unroll
  for (int r = 0; r < 8; r++) {
    const int row = m0 + r + ((lane & 16) ? 8 : 0);
    const int cc = n0 + col;
    if (row < M) C[(size_t)row * ldc + cc] = acc.f[r] + (bias ? bias[cc] : 0.0f);
  }
}

// ---------------------------------------------------------------------------
// Fused score pass: per batch, T[16,4096] = A9[16,256] @ em_K[b]^T where
// A9 rows 0..7 = top-8 candidate keys, row 8 = retrieve query q, rows 9..15 = 0.
// Reads em_K from HBM exactly once for both retrieve and write scoring.
// The 16x256 f32 key tile is staged to LDS with GLOBAL_LOAD_ASYNC_TO_LDS_B128
// (ASYNCcnt-tracked), padded rows (260 floats) for 16B alignment and
// conflict-free b64 fragment reads across all 64 banks.
// ---------------------------------------------------------------------------
constexpr int kTilePitch = 260;  // 256 + 4 pad floats (row stride = 1040B, 16B aligned)

__global__ void __launch_bounds__(32)
wmma_scores(const float* __restrict__ A9, const float* __restrict__ emK,
            const float* __restrict__ emS, float* __restrict__ SC,
            float* __restrict__ sret) {
#if HAVE_ASYNC_LDS
  __shared__ float tile[16 * kTilePitch];  // 16.6 KB key tile
#endif
  const int lane = threadIdx.x;
  const int b = blockIdx.y;
  const int m0 = blockIdx.x * 16;
  const int col = lane & 15;
  const int koff = (lane & 16) ? 2 : 0;

  const float* Ar = A9 + ((size_t)b * 16 + col) * kDem;
  const float* KbBase = emK + ((size_t)b * kM + m0) * kDem;

  union { v8f v; float f[8]; } acc;
#pragma unroll
  for (int i = 0; i < 8; i++) acc.f[i] = 0.0f;

#if HAVE_ASYNC_LDS
  // Stage the 16x256 tile: 32 async b128 transfers, 512B coalesced per wave.
  for (int ch = 0; ch < 32; ch++) {
    const int flat = ch * 32 + lane;  // units of 16B; 64 per key row
    const int row = flat >> 6;
    const int off16 = flat & 63;
    __builtin_amdgcn_global_load_async_to_lds_b128(
        TO_GLOBAL(KbBase + (size_t)row * kDem + off16 * 4),
        TO_LDS(&tile[row * kTilePitch + off16 * 4]), 0, 0);
  }
#if __has_builtin(__builtin_amdgcn_s_wait_asynccnt)
  __builtin_amdgcn_s_wait_asynccnt(0);
#else
  asm volatile("s_wait_asynccnt 0x0" ::: "memory");
#endif
  const float* Kl = &tile[col * kTilePitch + koff];
  const float* ap = Ar + koff;
  for (int k0 = 0; k0 < kDem; k0 += 4) {
    v2f a = *(const v2f*)(ap + k0);
    v2f bf = *(const v2f*)(Kl + k0);  // ds_load_b64, conflict-free
    acc.v = WMMA_F32(a, bf, acc.v);
  }
#else
  // Fallback: direct global B-fragment loads + prefetch down the key row.
  const float* Kb = KbBase + (size_t)col * kDem;
  for (int k0 = 0; k0 < kDem; k0 += 4) {
    v2f a = *(const v2f*)(Ar + k0 + koff);
    v2f bf = *(const v2f*)(Kb + k0 + koff);
    __builtin_prefetch(Kb + k0 + 128, 0, 0);
    acc.v = WMMA_F32(a, bf, acc.v);
  }
#endif

  const int m = m0 + col;
#pragma unroll
  for (int r = 0; r < 8; r++) {
    const int row = r + ((lane & 16) ? 8 : 0);
    if (row < 8) {
      SC[((size_t)b * kC + row) * kM + m] = acc.f[r];  // candidate c = row
    } else if (row == 8) {
      const float s = emS[(size_t)b * kM + m];
      sret[(size_t)b * kM + m] = (s > 0.0f) ? acc.f[r] : NEG_INF;
    }
    // rows 9..15 are zero-padding; discard.
  }
}

// ---------------------------------------------------------------------------
// Top-8 candidates per batch; gather K_C into A9 rows 0..7 (zero rows 9..15),
// V_C and top scores into workspace. Tie-break: lower index (jax top_k).
// ---------------------------------------------------------------------------
__global__ void __launch_bounds__(256)
topc_prep(const float* __restrict__ cscore, const float* __restrict__ candK,
          const float* __restrict__ candV, float* __restrict__ A9,
          float* __restrict__ VC, float* __restrict__ tcv) {
  __shared__ float sv[kP];
  __shared__ float rv[256];
  __shared__ int ri[256];
  __shared__ int selIdx[kC];
  __shared__ float selVal[kC];
  const int b = blockIdx.x, tid = threadIdx.x;

  for (int i = tid; i < kP; i += 256) sv[i] = cscore[(size_t)b * kP + i];
  __syncthreads();

  for (int c = 0; c < kC; c++) {
    float bv = NEG_INF; int bi = kP;
    for (int i = tid; i < kP; i += 256) {
      const float v = sv[i];
      if (v > bv || (v == bv && i < bi)) { bv = v; bi = i; }
    }
    rv[tid] = bv; ri[tid] = bi; __syncthreads();
    for (int s = 128; s > 0; s >>= 1) {
      if (tid < s) {
        if (rv[tid + s] > rv[tid] || (rv[tid + s] == rv[tid] && ri[tid + s] < ri[tid])) {
          rv[tid] = rv[tid + s]; ri[tid] = ri[tid + s];
        }
      }
      __syncthreads();
    }
    if (tid == 0) { selIdx[c] = ri[0]; selVal[c] = rv[0]; sv[ri[0]] = NEG_INF; }
    __syncthreads();
  }

  for (int c = 0; c < kC; c++) {
    const int idx = selIdx[c];
    A9[((size_t)b * 16 + c) * kDem + tid] = candK[((size_t)b * kP + idx) * kDem + tid];
    VC[((size_t)b * kC + c) * kDem + tid] = candV[((size_t)b * kP + idx) * kDem + tid];
  }
  for (int r = 9; r < 16; r++) A9[((size_t)b * 16 + r) * kDem + tid] = 0.0f;
  if (tid < kC) tcv[b * kC + tid] = selVal[tid];
}

// Unit-normalize q rows; also drop q into A9 row 8 for the fused score pass.
__global__ void __launch_bounds__(256)
normalize_q(const float* __restrict__ qpre, float* __restrict__ q, float* __restrict__ A9) {
  __shared__ float red[256];
  const int b = blockIdx.x, tid = threadIdx.x;
  const float v = qpre[(size_t)b * kDem + tid];
  red[tid] = v * v; __syncthreads();
  for (int s = 128; s > 0; s >>= 1) { if (tid < s) red[tid] += red[tid + s]; __syncthreads(); }
  const float out = v / fmaxf(sqrtf(red[0]), kEPS);
  q[(size_t)b * kDem + tid] = out;
  A9[((size_t)b * 16 + 8) * kDem + tid] = out;
}

// ---------------------------------------------------------------------------
// Retrieve: top-16 of masked scores, gather V_top, cross-attention, output
// 256-dim context (projected to D by the final WMMA GEMM).
// ---------------------------------------------------------------------------
__global__ void __launch_bounds__(256)
retrieve_attn(const float* __restrict__ sret, const float* __restrict__ emV,
              const float* __restrict__ qcross, float* __restrict__ retr) {
  __shared__ float sS[kM];            // 16 KB
  __shared__ float sV[kKRET * kDem];  // 16 KB
  __shared__ float rv[256];
  __shared__ int ri[256];
  __shared__ float tv[kKRET];
  __shared__ int tidx[kKRET];
  __shared__ float sw[kKRET];
  const int b = blockIdx.x, tid = threadIdx.x;

  for (int i = tid; i < kM; i += 256) sS[i] = sret[(size_t)b * kM + i];
  __syncthreads();

  for (int k = 0; k < kKRET; k++) {
    float bv = NEG_INF; int bi = kM;
    for (int i = tid; i < kM; i += 256) {
      const float v = sS[i];
      if (v > bv || (v == bv && i < bi)) { bv = v; bi = i; }
    }
    rv[tid] = bv; ri[tid] = bi; __syncthreads();
    for (int s = 128; s > 0; s >>= 1) {
      if (tid < s) {
        if (rv[tid + s] > rv[tid] || (rv[tid + s] == rv[tid] && ri[tid + s] < ri[tid])) {
          rv[tid] = rv[tid + s]; ri[tid] = ri[tid + s];
        }
      }
      __syncthreads();
    }
    if (tid == 0) { tv[k] = rv[0]; tidx[k] = ri[0]; sS[ri[0]] = NEG_INF; }
    __syncthreads();
  }

  for (int k = 0; k < kKRET; k++)
    sV[k * kDem + tid] = emV[((size_t)b * kM + tidx[k]) * kDem + tid];
  __syncthreads();

  if (tid < kKRET) {
    float a;
    if (tv[tid] == NEG_INF) {
      a = NEG_INF;
    } else {
      float acc = 0.0f;
      const float* qc = qcross + (size_t)b * kDem;
      for (int d = 0; d < kDem; d++) acc += qc[d] * sV[tid * kDem + d];
      a = acc * kCROSS;
    }
    rv[tid] = a;
  }
  __syncthreads();
  if (tid == 0) {
    float mx = NEG_INF;
    for (int k = 0; k < kKRET; k++) mx = fmaxf(mx, rv[k]);
    float sum = 0.0f;
    for (int k = 0; k < kKRET; k++) {
      const float e = (mx == NEG_INF) ? 0.0f : expf(rv[k] - mx);
      sw[k] = e; sum += e;
    }
    for (int k = 0; k < kKRET; k++) sw[k] = (sum > 0.0f) ? sw[k] / sum : 0.0f;  // nan_to_num
  }
  __syncthreads();

  float acc = 0.0f;
  for (int k = 0; k < kKRET; k++) acc += sw[k] * sV[k * kDem + tid];
  retr[(size_t)b * kDem + tid] = acc;
}

// Bulk float4 stream copy for outK/outV (dominant HBM traffic).
__global__ void bulk_copy4(const float4* __restrict__ src, float4* __restrict__ dst, size_t n) {
  size_t i = (size_t)blockIdx.x * blockDim.x + threadIdx.x;
  const size_t stride = (size_t)gridDim.x * blockDim.x;
  for (; i < n; i += stride) dst[i] = src[i];
}

// ---------------------------------------------------------------------------
// Sequential 8-step write scan, one workgroup per batch. S[4096] and the step
// score vector live in LDS; only modified slot rows (<=32) of K/V are
// materialized (workspace), with score fix-ups for slots whose K changed.
// Finally S is decayed + budget-enforced and modified rows are patched into
// outK/outV (bulk copies already ran earlier on this stream).
// ---------------------------------------------------------------------------
__global__ void __launch_bounds__(256)
scan_write(const float* __restrict__ SC, const float* __restrict__ A9,
           const float* __restrict__ VC, const float* __restrict__ tcv,
           const float* __restrict__ emS, const float* __restrict__ emK,
           const float* __restrict__ emV, const float* __restrict__ g_em,
           const int* __restrict__ wmask, float* __restrict__ modK,
           float* __restrict__ modV, float* __restrict__ outK,
           float* __restrict__ outV, float* __restrict__ outS) {
  __shared__ float sS[kM];  // current S   (16 KB)
  __shared__ float sc[kM];  // step scores (16 KB)
  __shared__ float rv[256];
  __shared__ int ri[256];
  __shared__ int modIds[kC * kKW];
  __shared__ int nmodSh;
  __shared__ float w4[kKW];
  __shared__ int id4[kKW];
  __shared__ int foundSh;
  const int b = blockIdx.x, tid = threadIdx.x;
  const float gate = (wmask[b] != 0) ? g_em[b] : 0.0f;

  for (int i = tid; i < kM; i += 256) sS[i] = emS[(size_t)b * kM + i];
  if (tid == 0) nmodSh = 0;
  __syncthreads();

  float* mK = modK + (size_t)b * (kC * kKW) * kDem;
  float* mV = modV + (size_t)b * (kC * kKW) * kDem;

  if (gate != 0.0f) {
    for (int c = 0; c < kC; c++) {
      const float* scc = SC + ((size_t)b * kC + c) * kM;
      const float* kc = A9 + ((size_t)b * 16 + c) * kDem;  // K_C row c
      const float* vc = VC + ((size_t)b * kC + c) * kDem;
      for (int i = tid; i < kM; i += 256) sc[i] = scc[i] - kWW * sS[i];
      __syncthreads();
      // Fix up slots whose K has been modified in previous steps.
      const int nmod = nmodSh;
      for (int i = tid; i < nmod; i += 256) {
        const float* kr = mK + (size_t)i * kDem;
        float d = 0.0f;
        for (int j = 0; j < kDem; j++) d += kc[j] * kr[j];
        sc[modIds[i]] = d - kWW * sS[modIds[i]];
      }
      __syncthreads();
      // Soft top-4 (sequential argmax, lower-index tie-break).
      float v4[kKW];
      for (int t = 0; t < kKW; t++) {
        float bv = NEG_INF; int bi = kM;
        for (int i = tid; i < kM; i += 256) {
          const float v = sc[i];
          if (v > bv || (v == bv && i < bi)) { bv = v; bi = i; }
        }
        rv[tid] = bv; ri[tid] = bi; __syncthreads();
        for (int s = 128; s > 0; s >>= 1) {
          if (tid < s) {
            if (rv[tid + s] > rv[tid] || (rv[tid + s] == rv[tid] && ri[tid + s] < ri[tid])) {
              rv[tid] = rv[tid + s]; ri[tid] = ri[tid + s];
            }
          }
          __syncthreads();
        }
        if (tid == 0) { id4[t] = ri[0]; sc[ri[0]] = NEG_INF; }
        v4[t] = rv[0];
        __syncthreads();
      }
      {  // softmax(vals / TAU); v4[0] is the max (sorted descending)
        float e[kKW], sum = 0.0f;
        for (int t = 0; t < kKW; t++) { e[t] = expf((v4[t] - v4[0]) / kTAU); sum += e[t]; }
        if (tid == 0) for (int t = 0; t < kKW; t++) w4[t] = e[t] / sum;
      }
      __syncthreads();
      const float s_c = tcv[b * kC + c];
      for (int t = 0; t < kKW; t++) {
        const int m = id4[t];
        const float a = w4[t] * gate;
        if (tid == 0) {
          int f = -1;
          const int n = nmodSh;
          for (int i = 0; i < n; i++) if (modIds[i] == m) { f = i; break; }
          if (f < 0) { f = n; modIds[f] = m; nmodSh = n + 1; foundSh = -(f + 1); }
          else foundSh = f + 1;
        }
        __syncthreads();
        const bool isNew = foundSh < 0;
        const int idx = isNew ? (-foundSh - 1) : (foundSh - 1);
        float* kr = mK + (size_t)idx * kDem;
        float* vr = mV + (size_t)idx * kDem;
        float kold, vold;
        if (isNew) {
          kold = emK[((size_t)b * kM + m) * kDem + tid];
          vold = emV[((size_t)b * kM + m) * kDem + tid];
        } else {
          kold = kr[tid]; vold = vr[tid];
        }
        const float kn = (1.0f - a) * kold + a * kc[tid];
        const float vn = (1.0f - a) * vold + a * vc[tid];
        rv[tid] = kn * kn; __syncthreads();
        for (int s = 128; s > 0; s >>= 1) { if (tid < s) rv[tid] += rv[tid + s]; __syncthreads(); }
        const float nrm = fmaxf(sqrtf(rv[0]), kEPS);
        kr[tid] = kn / nrm;  // unit_normalize of blended key
        vr[tid] = vn;
        if (tid == 0) sS[m] = fminf(fmaxf(sS[m] + a * s_c, 0.0f), kSMAX);
        __syncthreads();
      }
    }
  }

  // S: decay then budget-enforce.
  float part = 0.0f;
  for (int i = tid; i < kM; i += 256) part += sS[i] * kDECAY;
  rv[tid] = part; __syncthreads();
  for (int s = 128; s > 0; s >>= 1) { if (tid < s) rv[tid] += rv[tid + s]; __syncthreads(); }
  const float scale = fminf(1.0f, kBUDGET / fmaxf(rv[0], kEPS));
  for (int i = tid; i < kM; i += 256) outS[(size_t)b * kM + i] = sS[i] * kDECAY * scale;

  // Patch modified rows into the already-copied outK/outV.
  const int nmod = nmodSh;
  for (int i = 0; i < nmod; i++) {
    const int m = modIds[i];
    outK[((size_t)b * kM + m) * kDem + tid] = mK[(size_t)i * kDem + tid];
    outV[((size_t)b * kM + m) * kDem + tid] = mV[(size_t)i * kDem + tid];
  }
}

// ---------------------------------------------------------------------------
extern "C" void kernel_launch(void* const* d_in, const int* in_sizes, int n_in,
                              void* d_out, int out_size, void* d_ws, size_t ws_size,
                              hipStream_t stream) {
  (void)in_sizes; (void)n_in; (void)out_size; (void)ws_size;
  const float* x = (const float*)d_in[0];
  const float* y_wm = (const float*)d_in[1];
  const float* em_K = (const float*)d_in[2];
  const float* em_V = (const float*)d_in[3];
  const float* em_S = (const float*)d_in[4];
  const float* cand_K = (const float*)d_in[5];
  const float* cand_V = (const float*)d_in[6];
  const float* cand_score = (const float*)d_in[7];
  const int* write_mask = (const int*)d_in[8];
  const float* g_em = (const float*)d_in[9];
  const float* Wq_em_w = (const float*)d_in[10];
  const float* Wq_em_b = (const float*)d_in[11];
  const float* Wqc_w = (const float*)d_in[12];
  const float* Wqc_b = (const float*)d_in[13];
  const float* Wo_w = (const float*)d_in[14];
  const float* Wo_b = (const float*)d_in[15];

  float* y_em = (float*)d_out;                   // [32, 2048]
  float* outK = y_em + (size_t)kBS * kD;         // [32, 4096, 256]
  float* outV = outK + (size_t)kBS * kM * kDem;  // [32, 4096, 256]
  float* outS = outV + (size_t)kBS * kM * kDem;  // [32, 4096]

  float* ws = (float*)d_ws;  // ~7.8 MB total
  float* q = ws;      ws += (size_t)kBS * kDem;
  float* qcross = ws; ws += (size_t)kBS * kDem;
  float* qpre = ws;   ws += (size_t)kBS * kDem;
  float* A9 = ws;     ws += (size_t)kBS * 16 * kDem;
  float* VC = ws;     ws += (size_t)kBS * kC * kDem;
  float* tcv = ws;    ws += (size_t)kBS * kC;
  float* sret = ws;   ws += (size_t)kBS * kM;
  float* SC = ws;     ws += (size_t)kBS * kC * kM;
  float* retr = ws;   ws += (size_t)kBS * kDem;
  float* modK = ws;   ws += (size_t)kBS * (kC * kKW) * kDem;
  float* modV = ws;   ws += (size_t)kBS * (kC * kKW) * kDem;

  // 1) q_pre = concat(x, y_wm) @ Wq_em_w + b   (WMMA, K=4096 split at 2048)
  wmma_gemm_bias<<<dim3(kDem / 16, kBS / 16), 32, 0, stream>>>(
      x, y_wm, kD, kD, Wq_em_w, Wq_em_b, qpre, kDem, kBS, kDem, 2 * kD);
  // 2) q_cross = x @ Wqc_w + b                 (WMMA)
  wmma_gemm_bias<<<dim3(kDem / 16, kBS / 16), 32, 0, stream>>>(
      x, x, kD, kD, Wqc_w, Wqc_b, qcross, kDem, kBS, kDem, kD);
  // 3) top-8 candidates -> A9 rows 0..7, VC, tcv
  topc_prep<<<kBS, 256, 0, stream>>>(cand_score, cand_K, cand_V, A9, VC, tcv);
  // 4) q = unit_normalize(q_pre) -> A9 row 8
  normalize_q<<<kBS, 256, 0, stream>>>(qpre, q, A9);
  // 5) fused score pass: one em_K read feeds retrieve + write scoring (WMMA,
  //    async-staged key tiles in LDS)
  wmma_scores<<<dim3(kM / 16, kBS), 32, 0, stream>>>(A9, em_K, em_S, SC, sret);
  // 6) retrieve: top-16 + cross attention -> retr[32,256]
  retrieve_attn<<<kBS, 256, 0, stream>>>(sret, em_V, qcross, retr);
  // 7) y_em = retr @ Wo_w + Wo_b               (WMMA)
  wmma_gemm_bias<<<dim3(kD / 16, kBS / 16), 32, 0, stream>>>(
      retr, retr, kDem, kDem, Wo_w, Wo_b, y_em, kD, kBS, kD, kDem);
  // 8) bulk copy K/V to outputs (dominant HBM traffic)
  const size_t n4 = (size_t)kBS * kM * kDem / 4;
  bulk_copy4<<<8192, 256, 0, stream>>>((const float4*)em_K, (float4*)outK, n4);
  bulk_copy4<<<8192, 256, 0, stream>>>((const float4*)em_V, (float4*)outV, n4);
  // 9) sequential write scan; patches modified rows, emits outS
  scan_write<<<kBS, 256, 0, stream>>>(SC, A9, VC, tcv, em_S, em_K, em_V, g_em,
                                      write_mask, modK, modV, outK, outV, outS);
}